// WNGat_83047487635623
// MI455X (gfx1250) — compile-verified
//
#include <hip/hip_runtime.h>
#include <math.h>

typedef __attribute__((ext_vector_type(2))) float v2f;
typedef __attribute__((ext_vector_type(4))) float v4f;
typedef __attribute__((ext_vector_type(8))) float v8f;

#define HC 256          // HEADS * channels per layer output (4*64)
#define NHEADS 4
#define SLOPE 0.2f

// ---------------- float atomic max via int/uint punning -----------------
__device__ __forceinline__ void atomicMaxF(float* addr, float val) {
  if (val >= 0.0f) {
    atomicMax((int*)addr, __float_as_int(val));
  } else {
    atomicMin((unsigned int*)addr, __float_as_uint(val));
  }
}

// ---------------- fill ----------------
__global__ void fill_f32(float* __restrict__ p, float v, int n) {
  int i = blockIdx.x * blockDim.x + threadIdx.x;
  if (i < n) p[i] = v;
}

// ------- GEMM: H[N x 256] = X[N x K] * W[K x 256], wave-per-16x16-tile -------
// Uses V_WMMA_F32_16X16X4_F32 (fp32-native matrix core path on CDNA5).
__global__ void gemm_wmma_f32(const float* __restrict__ X, const float* __restrict__ W,
                              float* __restrict__ H, int K, int tilesTotal) {
  int wave = (int)((blockIdx.x * blockDim.x + threadIdx.x) >> 5);
  if (wave >= tilesTotal) return;           // whole wave uniform; active waves EXEC=all-1s
  int lane = threadIdx.x & 31;
  int tileM = wave >> 4;                    // 256 cols = 16 tiles of 16
  int tileN = wave & 15;
  int row0 = tileM << 4;
  int col0 = tileN << 4;
  int laneLo = lane & 15;                   // M (A) / N (B) index within tile
  int hi = lane >> 4;                       // 0 -> K{0,1}; 1 -> K{2,3}

  const float* xrow = X + (size_t)(row0 + laneLo) * (size_t)K;
  const float* wcol = W + col0 + laneLo;

  v8f acc = {};
  for (int k = 0; k < K; k += 4) {
    // A fragment: 16x4 f32 -> 2 VGPRs/lane (lanes 0-15: K=0,1; lanes 16-31: K=2,3)
    v2f a = *(const v2f*)(xrow + k + 2 * hi);
    // B fragment: 4x16 f32, row striped across lanes within a VGPR (mirrored layout)
    v2f b;
    b.x = wcol[(size_t)(k + 2 * hi) * HC];
    b.y = wcol[(size_t)(k + 2 * hi + 1) * HC];
    acc = __builtin_amdgcn_wmma_f32_16x16x4_f32(false, a, false, b, (short)0, acc,
                                                false, false);
  }
  // C/D layout: VGPR r -> M = r (lanes 0-15) / r+8 (lanes 16-31), N = lane%16
  float* hout = H + (size_t)row0 * HC + col0 + laneLo;
#pragma unroll
  for (int r = 0; r < 8; ++r) {
    hout[(size_t)(r + hi * 8) * HC] = acc[r];
  }
}

// ------- per-node attention logits: as[n,h] = <h[n,h,:], a_src[h,:]> -------
__global__ void alpha_kernel(const float* __restrict__ H, const float* __restrict__ a_src,
                             const float* __restrict__ a_dst,
                             float* __restrict__ as, float* __restrict__ ad, int n) {
  int wave = (int)((blockIdx.x * blockDim.x + threadIdx.x) >> 5);
  if (wave >= n) return;
  int lane = threadIdx.x & 31;
  int base = lane * 8;                      // 8 contiguous floats, all in same head
  const float* hp = H + (size_t)wave * HC + base;
  v4f h0 = *(const v4f*)(hp);
  v4f h1 = *(const v4f*)(hp + 4);
  v4f s0 = *(const v4f*)(a_src + base);
  v4f s1 = *(const v4f*)(a_src + base + 4);
  v4f d0 = *(const v4f*)(a_dst + base);
  v4f d1 = *(const v4f*)(a_dst + base + 4);
  float ps = h0.x * s0.x + h0.y * s0.y + h0.z * s0.z + h0.w * s0.w +
             h1.x * s1.x + h1.y * s1.y + h1.z * s1.z + h1.w * s1.w;
  float pd = h0.x * d0.x + h0.y * d0.y + h0.z * d0.z + h0.w * d0.w +
             h1.x * d1.x + h1.y * d1.y + h1.z * d1.z + h1.w * d1.w;
#pragma unroll
  for (int off = 1; off < 8; off <<= 1) {   // reduce within 8-lane head group
    ps += __shfl_xor(ps, off, 32);
    pd += __shfl_xor(pd, off, 32);
  }
  if ((lane & 7) == 0) {
    as[(size_t)wave * NHEADS + (lane >> 3)] = ps;
    ad[(size_t)wave * NHEADS + (lane >> 3)] = pd;
  }
}

__device__ __forceinline__ void edge_sd(const int* __restrict__ ei, int e, int E,
                                        int* s, int* d) {
  if (e < E) { *s = ei[e]; *d = ei[E + e]; }
  else       { *s = e - E; *d = e - E; }      // appended self-loops
}

// ------- edge pass 1: segment max of leaky_relu(as[src]+ad[dst]) over dst -------
__global__ void edge_max_kernel(const int* __restrict__ ei, int E, int Etot,
                                const float* __restrict__ as, const float* __restrict__ ad,
                                float* __restrict__ emax) {
  int e = blockIdx.x * blockDim.x + threadIdx.x;
  if (e >= Etot) return;
  int s, d; edge_sd(ei, e, E, &s, &d);
  v4f vs = *(const v4f*)(as + (size_t)s * NHEADS);
  v4f vd = *(const v4f*)(ad + (size_t)d * NHEADS);
#pragma unroll
  for (int h = 0; h < NHEADS; ++h) {
    float v = vs[h] + vd[h];
    v = v > 0.f ? v : SLOPE * v;
    atomicMaxF(&emax[(size_t)d * NHEADS + h], v);
  }
}

// ------- edge pass 2: ee = exp(e - emax[dst]); denom[dst] += ee -------
__global__ void edge_expsum_kernel(const int* __restrict__ ei, int E, int Etot,
                                   const float* __restrict__ as, const float* __restrict__ ad,
                                   const float* __restrict__ emax,
                                   float* __restrict__ ee, float* __restrict__ denom) {
  int e = blockIdx.x * blockDim.x + threadIdx.x;
  if (e >= Etot) return;
  int s, d; edge_sd(ei, e, E, &s, &d);
  v4f vs = *(const v4f*)(as + (size_t)s * NHEADS);
  v4f vd = *(const v4f*)(ad + (size_t)d * NHEADS);
  v4f vm = *(const v4f*)(emax + (size_t)d * NHEADS);
  v4f out;
#pragma unroll
  for (int h = 0; h < NHEADS; ++h) {
    float v = vs[h] + vd[h];
    v = v > 0.f ? v : SLOPE * v;
    float x = __expf(v - vm[h]);
    out[h] = x;
    atomicAdd(&denom[(size_t)d * NHEADS + h], x);
  }
  *(v4f*)(ee + (size_t)e * NHEADS) = out;
}

// ------- edge pass 3: acc[dst] += (ee/denom[dst]) * h[src]  (wave per edge) -------
__global__ void edge_aggr_kernel(const int* __restrict__ ei, int E, int Etot,
                                 const float* __restrict__ H, const float* __restrict__ ee,
                                 const float* __restrict__ denom, float* __restrict__ acc) {
  int wave = (int)((blockIdx.x * blockDim.x + threadIdx.x) >> 5);
  if (wave >= Etot) return;
  int lane = threadIdx.x & 31;
  int s, d; edge_sd(ei, wave, E, &s, &d);
  int head = lane >> 3;
  float coeff = ee[(size_t)wave * NHEADS + head] / denom[(size_t)d * NHEADS + head];
  int base = lane * 8;
  const float* hp = H + (size_t)s * HC + base;  // gather: h fits in 192MB L2
  v4f h0 = *(const v4f*)(hp);
  v4f h1 = *(const v4f*)(hp + 4);
  float* ap = acc + (size_t)d * HC + base;
  atomicAdd(ap + 0, coeff * h0.x);
  atomicAdd(ap + 1, coeff * h0.y);
  atomicAdd(ap + 2, coeff * h0.z);
  atomicAdd(ap + 3, coeff * h0.w);
  atomicAdd(ap + 4, coeff * h1.x);
  atomicAdd(ap + 5, coeff * h1.y);
  atomicAdd(ap + 6, coeff * h1.z);
  atomicAdd(ap + 7, coeff * h1.w);
}

// ------- epilogue: out = elu(acc + bias) -------
__global__ void bias_elu_kernel(const float* __restrict__ accv, const float* __restrict__ bias,
                                float* __restrict__ out, int n) {
  int i = blockIdx.x * blockDim.x + threadIdx.x;
  if (i >= n) return;
  float v = accv[i] + bias[i & (HC - 1)];
  out[i] = v > 0.f ? v : (__expf(v) - 1.f);
}

// ---------------------------------------------------------------------------
static void run_gat_layer(const float* x, int K, const float* W, const float* a_src,
                          const float* a_dst, const float* b, float* acc_out,
                          float* bufH, float* as, float* ad, float* emax, float* denom,
                          float* ee, const int* ei, int N, int E, hipStream_t stream) {
  const int Etot = E + N;
  const int tilesTotal = ((N + 15) / 16) * (HC / 16);
  gemm_wmma_f32<<<(tilesTotal + 7) / 8, 256, 0, stream>>>(x, W, bufH, K, tilesTotal);
  alpha_kernel<<<(N + 7) / 8, 256, 0, stream>>>(bufH, a_src, a_dst, as, ad, N);
  fill_f32<<<(N * NHEADS + 255) / 256, 256, 0, stream>>>(emax, -INFINITY, N * NHEADS);
  fill_f32<<<(N * NHEADS + 255) / 256, 256, 0, stream>>>(denom, 0.f, N * NHEADS);
  fill_f32<<<(N * HC + 255) / 256, 256, 0, stream>>>(acc_out, 0.f, N * HC);
  edge_max_kernel<<<(Etot + 255) / 256, 256, 0, stream>>>(ei, E, Etot, as, ad, emax);
  edge_expsum_kernel<<<(Etot + 255) / 256, 256, 0, stream>>>(ei, E, Etot, as, ad, emax,
                                                             ee, denom);
  edge_aggr_kernel<<<(Etot + 7) / 8, 256, 0, stream>>>(ei, E, Etot, bufH, ee, denom,
                                                       acc_out);
  bias_elu_kernel<<<(N * HC + 255) / 256, 256, 0, stream>>>(acc_out, b, acc_out, N * HC);
}

extern "C" void kernel_launch(void* const* d_in, const int* in_sizes, int n_in,
                              void* d_out, int out_size, void* d_ws, size_t ws_size,
                              hipStream_t stream) {
  const float* inp = (const float*)d_in[0];
  const int*   ei  = (const int*)d_in[1];
  const float* W1  = (const float*)d_in[2];
  const float* s1  = (const float*)d_in[3];
  const float* t1  = (const float*)d_in[4];
  const float* b1  = (const float*)d_in[5];
  const float* W2  = (const float*)d_in[6];
  const float* s2  = (const float*)d_in[7];
  const float* t2  = (const float*)d_in[8];
  const float* b2  = (const float*)d_in[9];
  const float* W3  = (const float*)d_in[10];
  const float* s3  = (const float*)d_in[11];
  const float* t3  = (const float*)d_in[12];
  const float* b3  = (const float*)d_in[13];

  const int IN = 128;
  const int N = in_sizes[0] / IN;      // 50000
  const int E = in_sizes[1] / 2;       // 800000
  const int Etot = E + N;

  // Workspace carve-up (~119 MB): h | activations | alphas | softmax state | per-edge exp
  float* bufH  = (float*)d_ws;                    // N*256
  float* buf1  = bufH  + (size_t)N * HC;          // N*256 (activations; aliased as acc)
  float* as    = buf1  + (size_t)N * HC;          // N*4
  float* ad    = as    + (size_t)N * NHEADS;      // N*4
  float* emax  = ad    + (size_t)N * NHEADS;      // N*4
  float* denom = emax  + (size_t)N * NHEADS;      // N*4
  float* ee    = denom + (size_t)N * NHEADS;      // Etot*4

  // Layer 1: x = inp (K=128) -> buf1
  run_gat_layer(inp, IN, W1, s1, t1, b1, buf1, bufH, as, ad, emax, denom, ee,
                ei, N, E, stream);
  // Layer 2: x = buf1 (K=256) -> buf1 (x only needed during GEMM; stream order makes
  // the in-place acc reuse safe)
  run_gat_layer(buf1, HC, W2, s2, t2, b2, buf1, bufH, as, ad, emax, denom, ee,
                ei, N, E, stream);
  // Layer 3: x = buf1 (K=256) -> d_out
  run_gat_layer(buf1, HC, W3, s3, t3, b3, (float*)d_out, bufH, as, ad, emax, denom, ee,
                ei, N, E, stream);
}